// EdgeEnhancementModule_58815282151958
// MI455X (gfx1250) — compile-verified
//
#include <hip/hip_runtime.h>
#include <hip/hip_bf16.h>
#include <math.h>

// Fused EdgeEnhancement kernel for MI455X (gfx1250, wave32).
// One pass: async-load x tile (40x40 with radius-4 halo) -> LDS,
// Sobel edges + 1x1 conv (7->4) via v_wmma_f32_16x16x4_f32 -> LDS,
// channel avg/max -> LDS, 7x7 conv + sigmoid, fused final multiply -> global.

typedef float v2f __attribute__((ext_vector_type(2)));
typedef float v8f __attribute__((ext_vector_type(8)));

#define TILE      32
#define XDIM      40            // TILE + 2*4 halo
#define ODIM      38            // TILE + 2*3 halo (region where 'out' is needed)
#define OPIX      (ODIM * ODIM) // 1444
#define OPAD      1456          // sOut row stride, >= max store index 1455
#define XCH       1600          // 40*40 per channel in LDS

__device__ __forceinline__ float sobel_edge(const float* sch, int cy, int cx) {
    // sch: channel base in x-LDS (stride XDIM), (cy,cx): center coords
    const float a00 = sch[(cy - 1) * XDIM + cx - 1];
    const float a01 = sch[(cy - 1) * XDIM + cx];
    const float a02 = sch[(cy - 1) * XDIM + cx + 1];
    const float a10 = sch[cy * XDIM + cx - 1];
    const float a12 = sch[cy * XDIM + cx + 1];
    const float a20 = sch[(cy + 1) * XDIM + cx - 1];
    const float a21 = sch[(cy + 1) * XDIM + cx];
    const float a22 = sch[(cy + 1) * XDIM + cx + 1];
    float gx = (a02 - a00) + 2.0f * (a12 - a10) + (a22 - a20);
    float gy = (a20 - a00) + 2.0f * (a21 - a01) + (a22 - a02);
    return __builtin_amdgcn_sqrtf(gx * gx + gy * gy); // raw v_sqrt_f32, ~1 ulp
}

__global__ __launch_bounds__(256) void edge_enh_fused(
    const float* __restrict__ x,     // (16,4,512,512)
    const float* __restrict__ w1,    // (4,7,1,1)
    const float* __restrict__ b1,    // (4,)
    const float* __restrict__ w_sa,  // (1,2,7,7)
    float* __restrict__ out)         // (16,4,512,512)
{
    __shared__ float sX[4 * XCH];      // x tile, 4 ch, 40x40, zero-padded halo
    __shared__ float sOut[4 * OPAD];   // 1x1-conv result on 38x38 region (padded)
    __shared__ float sAtt[2 * OPIX];   // [avg, max] on 38x38 region
    __shared__ float sWsa[98];         // 7x7x2 attention weights

    const int tid  = threadIdx.x;
    const int lane = tid & 31;
    const int wave = tid >> 5;
    const int n    = lane & 15;   // WMMA N index (output channel, valid < 4)
    const int half = lane >> 4;   // half-wave selects K rows per A/B VGPR

    const int batch = blockIdx.z;
    const int y0 = blockIdx.y * TILE;
    const int x0 = blockIdx.x * TILE;

    // ---- Stage 0: fill LDS x tile via async global->LDS; zeros for OOB ----
    {
        const unsigned long long gbase =
            (unsigned long long)(x + (size_t)batch * 4 * 512 * 512);
        const bool interior = (y0 >= 4) && (y0 + 36 <= 512) &&
                              (x0 >= 4) && (x0 + 36 <= 512);
        if (interior) {
            // whole tile in-bounds; 16B-aligned: b128 async copies
            for (int j = tid; j < 1600; j += 256) { // 4ch * 40rows * 10 segs
                const int c   = j / 400;
                const int rem = j - c * 400;
                const int yy  = rem / 10;
                const int xx  = (rem - yy * 10) << 2;
                const int gy  = y0 - 4 + yy;
                const int gx  = x0 - 4 + xx;
                const unsigned ldsaddr =
                    (unsigned)(unsigned long long)(void*)&sX[c * XCH + yy * XDIM + xx];
                const unsigned off = (unsigned)(((c << 18) + (gy << 9) + gx) << 2);
                asm volatile("global_load_async_to_lds_b128 %0, %1, %2"
                             :: "v"(ldsaddr), "v"(off), "s"(gbase)
                             : "memory");
            }
        } else {
            // boundary tile: per-element predicated b32 async, zeros for OOB
            for (int j = tid; j < 4 * XCH; j += 256) {
                const int c   = j / XCH;
                const int rem = j - c * XCH;
                const int yy  = rem / XDIM;
                const int xx  = rem - yy * XDIM;
                const int gy  = y0 - 4 + yy;
                const int gx  = x0 - 4 + xx;
                const unsigned ldsaddr = (unsigned)(unsigned long long)(void*)&sX[j];
                if (gy >= 0 && gy < 512 && gx >= 0 && gx < 512) {
                    const unsigned off = (unsigned)(((c << 18) + (gy << 9) + gx) << 2);
                    asm volatile("global_load_async_to_lds_b32 %0, %1, %2"
                                 :: "v"(ldsaddr), "v"(off), "s"(gbase)
                                 : "memory");
                } else {
                    sX[j] = 0.0f; // SAME-padding zeros
                }
            }
        }
        for (int j = tid; j < 98; j += 256) sWsa[j] = w_sa[j];
        asm volatile("s_wait_asynccnt 0x0" ::: "memory");
    }
    __syncthreads();

    // ---- Stage 1: 1x1 conv (7ch -> 4ch) over the 38x38 region via WMMA ----
    // D(16pix x 16n) = A1(16x4) x B1(4x16) + A2(16x4) x B2(4x16) + C(bias)
    // K chunk 1 = combined ch 0..3 (raw x), chunk 2 = ch 4..6 (Sobel) + pad.
    v2f bA, bB;
    const bool vn = (n < 4);
    bA.x = vn ? w1[n * 7 + 2 * half]     : 0.0f;  // B row K = 0 / 2
    bA.y = vn ? w1[n * 7 + 2 * half + 1] : 0.0f;  // B row K = 1 / 3
    bB.x = vn ? w1[n * 7 + 4 + 2 * half] : 0.0f;  // B row K = 4 / 6
    bB.y = (vn && half == 0) ? w1[n * 7 + 5] : 0.0f; // K = 5 / 7(pad)
    const float bias = vn ? b1[n] : 0.0f;

    for (int pb = wave * 16; pb < OPIX; pb += 8 * 16) { // wave-uniform loop
        int p = pb + n;
        if (p > OPIX - 1) p = OPIX - 1;          // clamp keeps EXEC all-ones
        const int py = p / ODIM;
        const int px = p - py * ODIM;
        const int cy = py + 1;                   // center coords in x-LDS
        const int cx = px + 1;

        v2f a1, a2;
        a1.x = sX[(2 * half) * XCH + cy * XDIM + cx];      // ch 0 / 2
        a1.y = sX[(2 * half + 1) * XCH + cy * XDIM + cx];  // ch 1 / 3
        a2.x = sobel_edge(&sX[(2 * half) * XCH], cy, cx);  // edge 0 / 2
        a2.y = (half == 0) ? sobel_edge(&sX[XCH], cy, cx) : 0.0f; // edge 1 / pad

        v8f acc;
#pragma unroll
        for (int r = 0; r < 8; ++r) acc[r] = bias;

        acc = __builtin_amdgcn_wmma_f32_16x16x4_f32(
            false, a1, false, bA, (short)0, acc, false, false);
        acc = __builtin_amdgcn_wmma_f32_16x16x4_f32(
            false, a2, false, bB, (short)0, acc, false, false);

        if (vn) { // lanes 0-3 and 16-19 hold valid output channels
            float* dst = &sOut[n * OPAD + pb + 8 * half]; // pix <= 1455 < OPAD
#pragma unroll
            for (int r = 0; r < 8; ++r) dst[r] = acc[r];  // D row M = r + 8*half
        }
    }
    __syncthreads();

    // ---- Stage 2: channel avg / max over 'out' on the 38x38 region ----
    for (int p = tid; p < OPIX; p += 256) {
        const float c0 = sOut[p];
        const float c1 = sOut[OPAD + p];
        const float c2 = sOut[2 * OPAD + p];
        const float c3 = sOut[3 * OPAD + p];
        sAtt[p]        = (c0 + c1 + c2 + c3) * 0.25f;
        sAtt[OPIX + p] = fmaxf(fmaxf(c0, c1), fmaxf(c2, c3));
    }
    __syncthreads();

    // ---- Stage 3: 7x7 attention conv + sigmoid + fused final multiply ----
    for (int q = tid; q < TILE * TILE; q += 256) {
        const int cy = q >> 5;
        const int cx = q & 31;
        float s = 0.0f;
#pragma unroll
        for (int ci = 0; ci < 2; ++ci) {
            const float* att = &sAtt[ci * OPIX];
            const float* wk  = &sWsa[ci * 49];
#pragma unroll
            for (int ky = 0; ky < 7; ++ky) {
                const float* row = &att[(cy + ky) * ODIM + cx];
#pragma unroll
                for (int kx = 0; kx < 7; ++kx)
                    s = fmaf(wk[ky * 7 + kx], row[kx], s);
            }
        }
        const float gain = 1.0f + 1.0f / (1.0f + __expf(-s));

        const int oidx = (cy + 3) * ODIM + (cx + 3); // center in 38x38 region
        const size_t gpix = (size_t)(y0 + cy) * 512 + (x0 + cx);
        const size_t base = ((size_t)batch * 4) * 512 * 512 + gpix;
#pragma unroll
        for (int co = 0; co < 4; ++co)
            out[base + (size_t)co * 512 * 512] = sOut[co * OPAD + oidx] * gain;
    }
}

extern "C" void kernel_launch(void* const* d_in, const int* in_sizes, int n_in,
                              void* d_out, int out_size, void* d_ws, size_t ws_size,
                              hipStream_t stream) {
    const float* x    = (const float*)d_in[0];
    const float* w1   = (const float*)d_in[1];
    const float* b1   = (const float*)d_in[2];
    const float* w_sa = (const float*)d_in[3];
    float* out = (float*)d_out;

    dim3 grid(512 / TILE, 512 / TILE, 16); // 16x16 tiles x 16 batches
    dim3 block(256);                       // 8 wave32 waves
    edge_enh_fused<<<grid, block, 0, stream>>>(x, w1, b1, w_sa, out);
}